// RegularizedFMNet_47270410060275
// MI455X (gfx1250) — compile-verified
//
#include <hip/hip_runtime.h>
#include <hip/hip_bf16.h>

typedef float v2f __attribute__((ext_vector_type(2)));
typedef float v8f __attribute__((ext_vector_type(8)));

#define B_  8
#define N_  10000
#define K_  128
#define C_  352
#define LAMBDA_ 0.001f

// ---------------------------------------------------------------------------
// Kernel 1: spectral projection  Out = E @ F
//   E:[K,N] (evecs_trans), F:[N,C] (feat)  ->  Out:[K,C]
// grid (2, 11, 2*B): block tile 64(M) x 32(C), N chunked by 64,
// double-buffered LDS. 128 threads = 4 waves; each wave owns TWO 16x16
// WMMA tiles (2x M register blocking, shared B fragment).
// F is staged TRANSPOSED in LDS so the B fragment is a contiguous pair
// (single ds_load_b64, no v_mov repacking).
// fp32 WMMA: v_wmma_f32_16x16x4_f32, K-step = 4.
// ---------------------------------------------------------------------------
__global__ __launch_bounds__(128) void proj_gemm(
    const float* __restrict__ feat_x, const float* __restrict__ feat_y,
    const float* __restrict__ etx,    const float* __restrict__ ety,
    float* __restrict__ Aout, float* __restrict__ Bout)
{
    __shared__ float Elds[2][64 * 65];   // [buf][64 M rows x 64 n], pad 65
    __shared__ float Flds[2][32 * 65];   // [buf][32 C rows x 64 n] (transposed)

    const int z     = blockIdx.z;
    const int b     = z >> 1;
    const int which = z & 1;
    const float* __restrict__ E = which ? ety    : etx;
    const float* __restrict__ F = which ? feat_y : feat_x;
    float*       __restrict__ O = which ? Bout   : Aout;

    const float* Eb = E + (size_t)b * K_ * N_;
    const float* Fb = F + (size_t)b * N_ * C_;
    float*       Ob = O + (size_t)b * K_ * C_;

    const int mbase = blockIdx.x * 64;
    const int cbase = blockIdx.y * 32;
    const int tid   = threadIdx.x;
    const int wave  = tid >> 5;
    const int lane  = tid & 31;
    const int l15   = lane & 15;
    const int hi    = lane >> 4;          // 0: K 0..1, 1: K 2..3 (frag striping)
    const int m0w   = (wave >> 1) * 16;   // wave M tiles: m0w and m0w+32
    const int c0w   = (wave & 1)  * 16;

    v8f acc0 = {0.f, 0.f, 0.f, 0.f, 0.f, 0.f, 0.f, 0.f};
    v8f acc1 = {0.f, 0.f, 0.f, 0.f, 0.f, 0.f, 0.f, 0.f};

    const int NCH = (N_ + 63) / 64;       // 157 chunks (last partial -> zero pad)

    // ---- staging lambda-equivalent (macro-free, inlined) ----
    // stage E (64x64) and F-transposed (32 C x 64 n) for chunk at nbase.
    auto stage = [&](int buf, int nbase) {
        for (int idx = tid; idx < 64 * 64; idx += 128) {
            const int r = idx >> 6, c = idx & 63;            // r = M row, c = n
            const int n = nbase + c;
            Elds[buf][r * 65 + c] =
                (n < N_) ? Eb[(size_t)(mbase + r) * N_ + n] : 0.f;
        }
        for (int idx = tid; idx < 64 * 32; idx += 128) {
            const int r = idx >> 5, c = idx & 31;            // r = n, c = C col
            const int n = nbase + r;
            const float v = (n < N_) ? Fb[(size_t)n * C_ + cbase + c] : 0.f;
            Flds[buf][c * 65 + r] = v;                       // transpose in LDS
        }
    };

    stage(0, 0);

    for (int ch = 0; ch < NCH; ++ch) {
        __syncthreads();                       // staging of chunk `ch` complete
        if (ch + 1 < NCH) stage((ch + 1) & 1, (ch + 1) * 64);

        const float* __restrict__ El = &Elds[ch & 1][0];
        const float* __restrict__ Fl = &Flds[ch & 1][0];

#pragma unroll
        for (int kk = 0; kk < 64; kk += 4) {
            v2f a0, a1, bf;
            // A frags (16x4): lanes0-15 -> K=kk..kk+1 ; lanes16-31 -> kk+2..kk+3
            a0.x = El[(m0w + l15)      * 65 + kk + 2 * hi];
            a0.y = El[(m0w + l15)      * 65 + kk + 2 * hi + 1];
            a1.x = El[(m0w + 32 + l15) * 65 + kk + 2 * hi];
            a1.y = El[(m0w + 32 + l15) * 65 + kk + 2 * hi + 1];
            // B frag (4x16) from transposed F: contiguous pair per lane
            bf.x = Fl[(c0w + l15) * 65 + kk + 2 * hi];
            bf.y = Fl[(c0w + l15) * 65 + kk + 2 * hi + 1];
            acc0 = __builtin_amdgcn_wmma_f32_16x16x4_f32(
                false, a0, false, bf, (short)0, acc0, false, false);
            acc1 = __builtin_amdgcn_wmma_f32_16x16x4_f32(
                false, a1, false, bf, (short)0, acc1, false, false);
        }
    }

    // D frag: VGPR r -> (M = r + 8*hi, N = l15)
#pragma unroll
    for (int r = 0; r < 8; ++r) {
        const int row = mbase + m0w + r + 8 * hi;
        const int col = cbase + c0w + l15;
        Ob[(size_t)row * C_ + col]        = acc0[r];
        Ob[(size_t)(row + 32) * C_ + col] = acc1[r];
    }
}

// ---------------------------------------------------------------------------
// Kernel 2: Gram matrices  G = X @ Y^T  over C=352 (88 WMMA K-steps).
// One wave per 16x16 output tile; operands are small & L2-resident.
// grid (64, 4*B), block = 32 (one wave).
// mat: 0 AA_t (A,A), 1 BB_t (B,B), 2 BA_t (B,A), 3 AB_t (A,B)
// ---------------------------------------------------------------------------
__global__ __launch_bounds__(32) void gram_wmma(
    const float* __restrict__ A, const float* __restrict__ Bm,
    float* __restrict__ AA, float* __restrict__ BB,
    float* __restrict__ BA, float* __restrict__ AB)
{
    const int tile = blockIdx.x;            // 0..63
    const int mat  = blockIdx.y & 3;
    const int b    = blockIdx.y >> 2;
    const int i0   = (tile >> 3) * 16;
    const int j0   = (tile & 7)  * 16;

    const float* X; const float* Y; float* G;
    if      (mat == 0) { X = A;  Y = A;  G = AA; }
    else if (mat == 1) { X = Bm; Y = Bm; G = BB; }
    else if (mat == 2) { X = Bm; Y = A;  G = BA; }
    else               { X = A;  Y = Bm; G = AB; }

    const float* Xb = X + (size_t)b * K_ * C_;
    const float* Yb = Y + (size_t)b * K_ * C_;

    const int lane = threadIdx.x & 31;
    const int l15  = lane & 15;
    const int hi   = lane >> 4;

    v8f acc = {0.f, 0.f, 0.f, 0.f, 0.f, 0.f, 0.f, 0.f};

    for (int c = 0; c < C_; c += 4) {
        v2f a, bf;
        a.x  = Xb[(size_t)(i0 + l15) * C_ + c + 2 * hi];
        a.y  = Xb[(size_t)(i0 + l15) * C_ + c + 2 * hi + 1];
        // B-matrix = Y^T, so Bmat[kred][j] = Y[j][c+kred]
        bf.x = Yb[(size_t)(j0 + l15) * C_ + c + 2 * hi];
        bf.y = Yb[(size_t)(j0 + l15) * C_ + c + 2 * hi + 1];
        acc = __builtin_amdgcn_wmma_f32_16x16x4_f32(
            false, a, false, bf, (short)0, acc, false, false);
    }

#pragma unroll
    for (int r = 0; r < 8; ++r) {
        const int row = i0 + r + 8 * hi;
        const int col = j0 + l15;
        G[(size_t)b * K_ * K_ + (size_t)row * K_ + col] = acc[r];
    }
}

// ---------------------------------------------------------------------------
// Kernel 3: resolvent masks D12, D21 ([B,K,K] each).
// evals are sorted ascending -> max is the last element.
// ---------------------------------------------------------------------------
__global__ __launch_bounds__(256) void resolvent_mask(
    const float* __restrict__ evx, const float* __restrict__ evy,
    float* __restrict__ D12, float* __restrict__ D21)
{
    const int idx = blockIdx.x * blockDim.x + threadIdx.x;
    if (idx >= 2 * B_ * K_ * K_) return;
    const int c   = idx & 127;
    const int r   = (idx >> 7) & 127;
    const int b   = (idx >> 14) & 7;
    const int dir = idx >> 17;                 // 0 -> D12, 1 -> D21

    const float* e1 = dir ? evy : evx;         // cols
    const float* e2 = dir ? evx : evy;         // rows

    const float scale = fmaxf(evx[b * K_ + (K_ - 1)], evy[b * K_ + (K_ - 1)]);
    const float a2 = sqrtf(e2[b * K_ + r] / scale);   // gamma = 0.5
    const float a1 = sqrtf(e1[b * K_ + c] / scale);
    const float d2 = a2 * a2 + 1.f, d1 = a1 * a1 + 1.f;
    const float re2 = a2 / d2, im2 = 1.f / d2;
    const float re1 = a1 / d1, im1 = 1.f / d1;
    const float dre = re2 - re1, dim = im2 - im1;

    (dir ? D21 : D12)[((size_t)b * K_ + r) * K_ + c] = dre * dre + dim * dim;
}

// ---------------------------------------------------------------------------
// Kernel 4: 2048 independent 128x128 SPD solves.
// One workgroup (128 threads) per (dir, b, i); augmented matrix in LDS
// (128 x 129 fp32 = 66 KB -- needs CDNA5's 320KB-per-WGP LDS).
// Unpivoted Gaussian elimination (G + lam*diag is SPD) + back substitution.
// ---------------------------------------------------------------------------
__global__ __launch_bounds__(128) void solve_rows(
    const float* __restrict__ AA, const float* __restrict__ BB,
    const float* __restrict__ BA, const float* __restrict__ AB,
    const float* __restrict__ D12, const float* __restrict__ D21,
    float* __restrict__ out)
{
    __shared__ float M[128 * 129];   // cols 0..127 = matrix, col 128 = rhs

    const int i   = blockIdx.x;
    const int b   = blockIdx.y;
    const int dir = blockIdx.z;
    const float* G = dir ? BB  : AA;
    const float* R = dir ? AB  : BA;
    const float* D = dir ? D21 : D12;
    const float* Gb = G + (size_t)b * K_ * K_;
    const int tid = threadIdx.x;

    // Build augmented system, coalesced global reads.
    for (int idx = tid; idx < K_ * K_; idx += 128) {
        const int r = idx >> 7, c = idx & 127;
        float v = Gb[idx];
        if (r == c) v += LAMBDA_ * D[((size_t)b * K_ + i) * K_ + c];
        M[r * 129 + c] = v;
    }
    M[tid * 129 + 128] = R[((size_t)b * K_ + i) * K_ + tid];

    // Forward elimination (row-parallel, stride-129 avoids bank conflicts).
    for (int k = 0; k < K_ - 1; ++k) {
        __syncthreads();
        const float inv = 1.0f / M[k * 129 + k];
        if (tid > k) {
            const float f = M[tid * 129 + k] * inv;
            const float* __restrict__ prow = &M[k * 129];
            float* __restrict__ trow = &M[tid * 129];
            for (int c = k + 1; c <= 128; ++c)
                trow[c] -= f * prow[c];
        }
    }

    // Back substitution.
    for (int k = K_ - 1; k >= 0; --k) {
        __syncthreads();
        const float xk = M[k * 129 + 128] / M[k * 129 + k];
        __syncthreads();
        if (tid < k)  M[tid * 129 + 128] -= M[tid * 129 + k] * xk;
        if (tid == k) M[k * 129 + 128] = xk;
    }
    __syncthreads();

    out[(((size_t)dir * B_ + b) * K_ + i) * K_ + tid] = M[tid * 129 + 128];
}

// ---------------------------------------------------------------------------
extern "C" void kernel_launch(void* const* d_in, const int* in_sizes, int n_in,
                              void* d_out, int out_size, void* d_ws, size_t ws_size,
                              hipStream_t stream)
{
    const float* feat_x  = (const float*)d_in[0];
    const float* feat_y  = (const float*)d_in[1];
    const float* evals_x = (const float*)d_in[2];
    const float* evals_y = (const float*)d_in[3];
    const float* etx     = (const float*)d_in[4];
    const float* ety     = (const float*)d_in[5];
    float* out = (float*)d_out;

    // Workspace layout (floats): A, Bm, AA, BB, BA, AB, D12, D21  (~6.0 MB)
    float* ws  = (float*)d_ws;
    float* A   = ws;
    float* Bm  = A   + (size_t)B_ * K_ * C_;
    float* AA  = Bm  + (size_t)B_ * K_ * C_;
    float* BB  = AA  + (size_t)B_ * K_ * K_;
    float* BA  = BB  + (size_t)B_ * K_ * K_;
    float* AB  = BA  + (size_t)B_ * K_ * K_;
    float* D12 = AB  + (size_t)B_ * K_ * K_;
    float* D21 = D12 + (size_t)B_ * K_ * K_;

    dim3 g1(K_ / 64, C_ / 32, 2 * B_);            // (2, 11, 16)
    proj_gemm<<<g1, 128, 0, stream>>>(feat_x, feat_y, etx, ety, A, Bm);

    dim3 g2(64, 4 * B_);
    gram_wmma<<<g2, 32, 0, stream>>>(A, Bm, AA, BB, BA, AB);

    resolvent_mask<<<(2 * B_ * K_ * K_ + 255) / 256, 256, 0, stream>>>(
        evals_x, evals_y, D12, D21);

    dim3 g4(K_, B_, 2);
    solve_rows<<<g4, 128, 0, stream>>>(AA, BB, BA, AB, D12, D21, out);
}